// GAE_65618510348475
// MI455X (gfx1250) — compile-verified
//
#include <hip/hip_runtime.h>
#include <hip/hip_bf16.h>

// ---------------------------------------------------------------------------
// GCN autoencoder for MI455X (gfx1250, wave32).
// GEMMs run on the WMMA pipe via V_WMMA_F32_16X16X4_F32 (fp32-exact, matches
// the fp32 reference). Aggregations are edge-parallel float4 scatters with
// global_atomic_add_f32 (memory-bound path; ~4 GB traffic @ 23.3 TB/s).
// ---------------------------------------------------------------------------

typedef float v2f __attribute__((ext_vector_type(2)));
typedef float v8f __attribute__((ext_vector_type(8)));

__device__ __forceinline__ v8f wmma_f32_16x16x4(v2f a, v2f b, v8f c) {
  // 8 args: (neg_a, A, neg_b, B, c_mod, C, reuse_a, reuse_b)
  return __builtin_amdgcn_wmma_f32_16x16x4_f32(false, a, false, b, (short)0, c,
                                               false, false);
}

// ---------------------------------------------------------------------------
// WMMA GEMM: C[M x Nc] = A[M x K] * B
//   OPB == 1 : B element (k,j) = B[j*ldb + k]   (i.e. C = A @ W^T, W row-major)
//   OPB == 0 : B element (k,j) = B[k*ldb + j]   (i.e. C = A @ W,   W row-major)
// Block tile 128x128, 8 waves (256 thr), wave tile 64x32 (4x2 WMMA tiles),
// K staged through LDS in steps of 16. Requires K % 16 == 0, Nc % 128 == 0.
// ---------------------------------------------------------------------------
template <int OPB>
__global__ void __launch_bounds__(256)
gemm_wmma(const float* __restrict__ A, int lda,
          const float* __restrict__ B, int ldb,
          float* __restrict__ C, int ldc,
          int M, int Nc, int K) {
  constexpr int BM = 128, BN = 128, BK = 16;
  constexpr int ALD = 20;   // padded A row stride (floats); 20*4 = 80 B (16B aligned)
  constexpr int BLD = 132;  // padded B row stride (floats); 132*4 = 528 B (16B aligned)

  __shared__ float As[BM * ALD];
  __shared__ float Bs[BK * BLD];

  const int tid  = threadIdx.x;
  const int wid  = tid >> 5;        // wave 0..7
  const int lane = tid & 31;
  const int lh   = lane >> 4;       // K-half select (0/1)
  const int lm   = lane & 15;       // M (A) / N (B) index within tile

  const int m0 = blockIdx.y * BM;
  const int n0 = blockIdx.x * BN;
  const int wm = (wid & 1) * 64;    // wave row offset in block
  const int wn = (wid >> 1) * 32;   // wave col offset in block

  v8f acc[4][2];
#pragma unroll
  for (int tm = 0; tm < 4; ++tm)
#pragma unroll
    for (int tn = 0; tn < 2; ++tn) acc[tm][tn] = (v8f)0.0f;

  for (int k0 = 0; k0 < K; k0 += BK) {
    __syncthreads();
    // ---- stage A tile [m0..m0+128) x [k0..k0+16) into LDS (2 x b128/thread)
#pragma unroll
    for (int r = 0; r < 2; ++r) {
      const int i   = tid + r * 256;
      const int row = i >> 2;
      const int c4  = (i & 3) << 2;
      const int gm  = m0 + row;
      float4 v = make_float4(0.f, 0.f, 0.f, 0.f);
      if (gm < M)
        v = *(const float4*)(A + (size_t)gm * lda + k0 + c4);
      *(float4*)(&As[row * ALD + c4]) = v;
    }
    // ---- stage B tile [k0..k0+16) x [n0..n0+128) into LDS
    if (OPB) {
      // B stored [Nc x K] row-major: read a j-row run of 16 k's, transpose in.
#pragma unroll
      for (int r = 0; r < 2; ++r) {
        const int i  = tid + r * 256;
        const int j  = i >> 2;
        const int k4 = (i & 3) << 2;
        const int gj = n0 + j;
        float4 v = make_float4(0.f, 0.f, 0.f, 0.f);
        if (gj < Nc)
          v = *(const float4*)(B + (size_t)gj * ldb + k0 + k4);
        Bs[(k4 + 0) * BLD + j] = v.x;
        Bs[(k4 + 1) * BLD + j] = v.y;
        Bs[(k4 + 2) * BLD + j] = v.z;
        Bs[(k4 + 3) * BLD + j] = v.w;
      }
    } else {
      // B stored [K x Nc] row-major: direct b128 copy.
#pragma unroll
      for (int r = 0; r < 2; ++r) {
        const int i  = tid + r * 256;
        const int kk = i >> 5;
        const int c  = (i & 31) << 2;
        const int gj = n0 + c;
        float4 v = make_float4(0.f, 0.f, 0.f, 0.f);
        if (gj < Nc)
          v = *(const float4*)(B + (size_t)(k0 + kk) * ldb + gj);
        *(float4*)(&Bs[kk * BLD + c]) = v;
      }
    }
    // ---- prefetch next A stage (speculative, global_prefetch_b8)
    if (k0 + BK < K) {
      const int gm = m0 + (tid >> 1);
      if (gm < M)
        __builtin_prefetch(A + (size_t)gm * lda + (k0 + BK) + (tid & 1) * 8, 0, 1);
    }
    __syncthreads();

    // ---- compute: 4 k-steps of 4, 4x2 WMMA tiles per wave
#pragma unroll
    for (int kk = 0; kk < BK; kk += 4) {
      const int ks = kk + 2 * lh;     // this lane-half's K pair base
      v2f a[4], b[2];
#pragma unroll
      for (int tm = 0; tm < 4; ++tm) {
        // A frag: row = wm + tm*16 + lm, K = ks, ks+1 (8B-aligned ds_load_b64)
        a[tm] = *(const v2f*)(&As[(wm + tm * 16 + lm) * ALD + ks]);
      }
#pragma unroll
      for (int tn = 0; tn < 2; ++tn) {
        const int col = wn + tn * 16 + lm;
        b[tn].x = Bs[(ks + 0) * BLD + col];
        b[tn].y = Bs[(ks + 1) * BLD + col];
      }
#pragma unroll
      for (int tm = 0; tm < 4; ++tm)
#pragma unroll
        for (int tn = 0; tn < 2; ++tn)
          acc[tm][tn] = wmma_f32_16x16x4(a[tm], b[tn], acc[tm][tn]);
    }
  }

  // ---- store C: VGPR v holds (M = v + 8*lh, N = lm) of each 16x16 tile
#pragma unroll
  for (int tm = 0; tm < 4; ++tm) {
#pragma unroll
    for (int v = 0; v < 8; ++v) {
      const int gm = m0 + wm + tm * 16 + v + 8 * lh;
      if (gm < M) {
#pragma unroll
        for (int tn = 0; tn < 2; ++tn) {
          const int gn = n0 + wn + tn * 16 + lm;
          C[(size_t)gm * ldc + gn] = acc[tm][tn][v];
        }
      }
    }
  }
}

// ---------------------------------------------------------------------------
// Graph-norm precompute
// ---------------------------------------------------------------------------
__global__ void init_deg_kernel(float* deg, int n) {
  int i = blockIdx.x * blockDim.x + threadIdx.x;
  if (i < n) deg[i] = 1.0f;  // self-loop contributes 1 to every node
}

__global__ void deg_accum_kernel(const int* __restrict__ dst, float* deg, int E) {
  int e = blockIdx.x * blockDim.x + threadIdx.x;
  if (e < E) atomicAdd(&deg[dst[e]], 1.0f);
}

__global__ void dinv_kernel(float* deg, int n) {
  int i = blockIdx.x * blockDim.x + threadIdx.x;
  if (i < n) {
    float d = deg[i];
    deg[i] = (d > 0.f) ? rsqrtf(d) : 0.f;
  }
}

__global__ void norm_kernel(const int* __restrict__ src, const int* __restrict__ dst,
                            const float* __restrict__ dinv, float* __restrict__ norm,
                            int E, int Et) {
  int e = blockIdx.x * blockDim.x + threadIdx.x;
  if (e >= Et) return;
  if (e < E) {
    norm[e] = dinv[src[e]] * dinv[dst[e]];
  } else {
    float d = dinv[e - E];
    norm[e] = d * d;
  }
}

// out[n][d] = b[d]  (bias broadcast, float4-vectorized)
__global__ void bias_init_kernel(float* __restrict__ out, const float* __restrict__ b,
                                 int n, int D4) {
  int i = blockIdx.x * blockDim.x + threadIdx.x;
  if (i >= n * D4) return;
  const int j = i % D4;
  ((float4*)out)[i] = ((const float4*)b)[j];
}

// out[dst[e]] += h[src[e]] * norm[e]  (edge-parallel, float4 per thread)
__global__ void scatter_kernel(const float* __restrict__ h,
                               const int* __restrict__ src, const int* __restrict__ dst,
                               const float* __restrict__ norm, float* __restrict__ out,
                               int D, int D4, int E, int Et) {
  long long tid = (long long)blockIdx.x * blockDim.x + threadIdx.x;
  long long total = (long long)Et * D4;
  if (tid >= total) return;
  const int e = (int)(tid / D4);
  const int j = (int)(tid % D4);
  int s, d;
  if (e < E) { s = src[e]; d = dst[e]; }
  else       { s = e - E;  d = e - E;  }
  const float nv = norm[e];
  const float4 v = ((const float4*)(h + (size_t)s * D))[j];
  float* o = out + (size_t)d * D + j * 4;
  atomicAdd(o + 0, v.x * nv);
  atomicAdd(o + 1, v.y * nv);
  atomicAdd(o + 2, v.z * nv);
  atomicAdd(o + 3, v.w * nv);
}

// ---------------------------------------------------------------------------
// Host-side orchestration
// ---------------------------------------------------------------------------
static inline void launch_gemm(int opb, const float* A, int lda, const float* B,
                               int ldb, float* C, int ldc, int M, int Nc, int K,
                               hipStream_t stream) {
  dim3 grid(Nc / 128, (M + 127) / 128);
  if (opb)
    gemm_wmma<1><<<grid, 256, 0, stream>>>(A, lda, B, ldb, C, ldc, M, Nc, K);
  else
    gemm_wmma<0><<<grid, 256, 0, stream>>>(A, lda, B, ldb, C, ldc, M, Nc, K);
}

extern "C" void kernel_launch(void* const* d_in, const int* in_sizes, int n_in,
                              void* d_out, int out_size, void* d_ws, size_t ws_size,
                              hipStream_t stream) {
  const int IN = 1024, HID = 512, OUT = 128;

  const float* features = (const float*)d_in[0];
  const int*   edge     = (const int*)d_in[1];   // [2, E] row-major
  const float* W1       = (const float*)d_in[2]; // [HID, IN]
  const float* b1       = (const float*)d_in[3];
  const float* W2       = (const float*)d_in[4]; // [OUT, HID]
  const float* b2       = (const float*)d_in[5];
  const float* b3       = (const float*)d_in[6];
  const float* b4       = (const float*)d_in[7];
  const float* head1    = (const float*)d_in[8]; // [OUT, OUT]

  const int n  = in_sizes[0] / IN;   // 20000
  const int E  = in_sizes[1] / 2;    // 160000
  const int Et = E + n;              // edges + self loops
  const int* src = edge;
  const int* dst = edge + E;

  float* out = (float*)d_out;
  float* z   = out;                       // [n, OUT]
  float* h2  = out + (size_t)n * OUT;     // [n, OUT]
  float* h4  = out + (size_t)n * OUT * 2; // [n, IN]

  // workspace layout (floats): dinv[n] | norm[Et] | t0[n*HID] | t1[n*HID]
  //                            | h2p[n*OUT] | t2[n*IN]   (~175 MB total)
  float* ws   = (float*)d_ws;
  float* dinv = ws;
  float* norm = dinv + n;
  float* t0   = norm + Et;
  float* t1   = t0 + (size_t)n * HID;
  float* h2p  = t1 + (size_t)n * HID;
  float* t2   = h2p + (size_t)n * OUT;

  const int TB = 256;
  // 1) symmetric normalization coefficients
  init_deg_kernel<<<(n + TB - 1) / TB, TB, 0, stream>>>(dinv, n);
  deg_accum_kernel<<<(E + TB - 1) / TB, TB, 0, stream>>>(dst, dinv, E);
  dinv_kernel<<<(n + TB - 1) / TB, TB, 0, stream>>>(dinv, n);
  norm_kernel<<<(Et + TB - 1) / TB, TB, 0, stream>>>(src, dst, dinv, norm, E, Et);

  // 2) conv1: h1 = agg(features @ W1^T) + b1
  launch_gemm(1, features, IN, W1, IN, t0, HID, n, HID, IN, stream);
  bias_init_kernel<<<((n * (HID / 4)) + TB - 1) / TB, TB, 0, stream>>>(t1, b1, n, HID / 4);
  {
    long long tot = (long long)Et * (HID / 4);
    scatter_kernel<<<(unsigned)((tot + TB - 1) / TB), TB, 0, stream>>>(
        t0, src, dst, norm, t1, HID, HID / 4, E, Et);
  }

  // 3) conv2: h2 = agg(h1 @ W2^T) + b2
  launch_gemm(1, t1, HID, W2, HID, h2p, OUT, n, OUT, HID, stream);
  bias_init_kernel<<<((n * (OUT / 4)) + TB - 1) / TB, TB, 0, stream>>>(h2, b2, n, OUT / 4);
  {
    long long tot = (long long)Et * (OUT / 4);
    scatter_kernel<<<(unsigned)((tot + TB - 1) / TB), TB, 0, stream>>>(
        h2p, src, dst, norm, h2, OUT, OUT / 4, E, Et);
  }

  // 4) head: z = h2 @ head1
  launch_gemm(0, h2, OUT, head1, OUT, z, OUT, n, OUT, OUT, stream);

  // 5) conv3 (tied W2^T): h3 = agg(z @ W2) + b3
  launch_gemm(0, z, OUT, W2, HID, t0, HID, n, HID, OUT, stream);
  bias_init_kernel<<<((n * (HID / 4)) + TB - 1) / TB, TB, 0, stream>>>(t1, b3, n, HID / 4);
  {
    long long tot = (long long)Et * (HID / 4);
    scatter_kernel<<<(unsigned)((tot + TB - 1) / TB), TB, 0, stream>>>(
        t0, src, dst, norm, t1, HID, HID / 4, E, Et);
  }

  // 6) conv4 (tied W1^T): h4 = agg(h3 @ W1) + b4
  launch_gemm(0, t1, HID, W1, IN, t2, IN, n, IN, HID, stream);
  bias_init_kernel<<<((n * (IN / 4)) + TB - 1) / TB, TB, 0, stream>>>(h4, b4, n, IN / 4);
  {
    long long tot = (long long)Et * (IN / 4);
    scatter_kernel<<<(unsigned)((tot + TB - 1) / TB), TB, 0, stream>>>(
        t2, src, dst, norm, h4, IN, IN / 4, E, Et);
  }
}